// RewireAttODEblock_79096117723815
// MI455X (gfx1250) — compile-verified
//
#include <hip/hip_runtime.h>
#include <math.h>
#include <stddef.h>

typedef float v2f __attribute__((ext_vector_type(2)));
typedef float v8f __attribute__((ext_vector_type(8)));

#define THREADS 256

// ---------- helpers ----------
__device__ __forceinline__ unsigned f2ord(float f) {
  unsigned u = __float_as_uint(f);
  return (u & 0x80000000u) ? ~u : (u | 0x80000000u);
}
__device__ __forceinline__ float ord2f(unsigned u) {
  unsigned b = (u & 0x80000000u) ? (u & 0x7FFFFFFFu) : ~u;
  return __uint_as_float(b);
}
__device__ __forceinline__ void atomAddF(float* p, float v) {
  __hip_atomic_fetch_add(p, v, __ATOMIC_RELAXED, __HIP_MEMORY_SCOPE_AGENT);
}
__device__ __forceinline__ void atomMaxU(unsigned* p, unsigned v) {
  __hip_atomic_fetch_max(p, v, __ATOMIC_RELAXED, __HIP_MEMORY_SCOPE_AGENT);
}
__device__ __forceinline__ void atomAddU(unsigned* p, unsigned v) {
  __hip_atomic_fetch_add(p, v, __ATOMIC_RELAXED, __HIP_MEMORY_SCOPE_AGENT);
}

// ---------- generic fill / copy ----------
__global__ void fill_u32_kernel(unsigned* p, unsigned v, size_t n) {
  size_t i = (size_t)blockIdx.x * blockDim.x + threadIdx.x;
  if (i < n) p[i] = v;
}
__global__ void copy_f32_kernel(float* dst, const float* src, size_t n) {
  size_t i = (size_t)blockIdx.x * blockDim.x + threadIdx.x;
  if (i < n) dst[i] = src[i];
}

// ---------- projections: q = x Wq, k = x Wk via V_WMMA_F32_16X16X4_F32 ----------
// One wave per 16-row tile of x; 4 n-tiles for Wq + 4 for Wk; K=64 in 16 steps of 4.
__global__ void proj_wmma_kernel(const float* __restrict__ x,
                                 const float* __restrict__ Wq,
                                 const float* __restrict__ Wk,
                                 float* __restrict__ q,
                                 float* __restrict__ kf,
                                 int ntiles) {
  int gwave = (int)((blockIdx.x * blockDim.x + threadIdx.x) >> 5);
  if (gwave >= ntiles) return;                 // wave-uniform: EXEC stays all-ones in WMMA region
  int lane = threadIdx.x & 31;
  int m16 = lane & 15;                         // M for A, N for B/D
  int ks = (lane >> 4) << 1;                   // K sub-pair: 0 or 2

  const float* xrow = x + ((size_t)gwave * 16 + m16) * 64;

  v8f acc[8];
  #pragma unroll
  for (int i = 0; i < 8; ++i)
    #pragma unroll
    for (int j = 0; j < 8; ++j) acc[i][j] = 0.0f;

  for (int k0 = 0; k0 < 64; k0 += 4) {
    v2f a;
    a.x = xrow[k0 + ks];
    a.y = xrow[k0 + ks + 1];
    #pragma unroll
    for (int t = 0; t < 4; ++t) {
      const float* wq0 = Wq + (size_t)(k0 + ks) * 64 + t * 16 + m16;
      const float* wk0 = Wk + (size_t)(k0 + ks) * 64 + t * 16 + m16;
      v2f bq, bk;
      bq.x = wq0[0];  bq.y = wq0[64];
      bk.x = wk0[0];  bk.y = wk0[64];
      acc[t]     = __builtin_amdgcn_wmma_f32_16x16x4_f32(false, a, false, bq,
                        (short)0, acc[t], false, false);
      acc[4 + t] = __builtin_amdgcn_wmma_f32_16x16x4_f32(false, a, false, bk,
                        (short)0, acc[4 + t], false, false);
    }
  }
  int mh = (lane >> 4) * 8;                    // C/D: VGPR v -> M = v + 8*(lane/16)
  #pragma unroll
  for (int t = 0; t < 4; ++t) {
    #pragma unroll
    for (int v = 0; v < 8; ++v) {
      size_t ri = ((size_t)gwave * 16 + mh + v) * 64 + (size_t)(t * 16 + m16);
      q[ri]  = acc[t][v];
      kf[ri] = acc[4 + t][v];
    }
  }
}

// ---------- per-edge attention scores [E,4] ----------
__global__ void score_kernel(const float* __restrict__ q, const float* __restrict__ kf,
                             const int* __restrict__ row, const int* __restrict__ col,
                             float* __restrict__ scores, int E) {
  int e = blockIdx.x * blockDim.x + threadIdx.x;
  if (e >= E) return;
  const float4* qp = (const float4*)(q + (size_t)row[e] * 64);
  const float4* kp = (const float4*)(kf + (size_t)col[e] * 64);
  #pragma unroll
  for (int h = 0; h < 4; ++h) {
    float s = 0.0f;
    #pragma unroll
    for (int i = 0; i < 4; ++i) {
      float4 a = qp[h * 4 + i];
      float4 b = kp[h * 4 + i];
      s += a.x * b.x + a.y * b.y + a.z * b.z + a.w * b.w;
    }
    scores[(size_t)e * 4 + h] = s * 0.25f;    // / sqrt(DK), DK=16
  }
}

// ---------- segment max over row (order-preserving uint atomicMax) ----------
__global__ void segmax_kernel(const float* scores, const int* row, unsigned* smax,
                              const float* meanatt, const float* thr, int use_mask, int E) {
  int e = blockIdx.x * blockDim.x + threadIdx.x;
  if (e >= E) return;
  bool m = true;
  if (use_mask) m = meanatt[e] > thr[0];
  int r = row[e];
  #pragma unroll
  for (int h = 0; h < 4; ++h) {
    float s = m ? scores[(size_t)e * 4 + h] : -1e9f;
    atomMaxU(&smax[r * 4 + h], f2ord(s));
  }
}

// ---------- segment sum of exp(score - max) ----------
__global__ void expsum_kernel(const float* scores, const unsigned* smax, const int* row,
                              float* ssum, const float* meanatt, const float* thr,
                              int use_mask, int E) {
  int e = blockIdx.x * blockDim.x + threadIdx.x;
  if (e >= E) return;
  bool m = true;
  if (use_mask) m = meanatt[e] > thr[0];
  int r = row[e];
  #pragma unroll
  for (int h = 0; h < 4; ++h) {
    float s = m ? scores[(size_t)e * 4 + h] : -1e9f;
    float mx = ord2f(smax[r * 4 + h]);
    atomAddF(&ssum[r * 4 + h], expf(s - mx));
  }
}

// ---------- per-edge mean attention (pass1: mean_att0, pass2: w) ----------
__global__ void attw_kernel(const float* scores, const unsigned* smax, const float* ssum,
                            const int* row, const float* meanatt, const float* thr,
                            int use_mask, float* outw, int E) {
  int e = blockIdx.x * blockDim.x + threadIdx.x;
  if (e >= E) return;
  if (use_mask && !(meanatt[e] > thr[0])) { outw[e] = 0.0f; return; }
  int r = row[e];
  float a = 0.0f;
  #pragma unroll
  for (int h = 0; h < 4; ++h) {
    float s = scores[(size_t)e * 4 + h];
    float mx = ord2f(smax[r * 4 + h]);
    a += expf(s - mx) / (ssum[r * 4 + h] + 1e-16f);
  }
  outw[e] = 0.25f * a;
}

// ---------- device-side quantile: 2-level 16-bit radix select ----------
__global__ void hist_hi_kernel(const float* vals, unsigned* hist, int E) {
  int e = blockIdx.x * blockDim.x + threadIdx.x;
  if (e >= E) return;
  unsigned key = f2ord(vals[e]);
  atomAddU(&hist[key >> 16], 1u);
}
__global__ void pick_hi_kernel(const unsigned* hist, unsigned r0, unsigned r1,
                               unsigned* selhi, unsigned* selrem) {
  unsigned ranks[2] = {r0, r1};
  unsigned long long cum = 0;
  int found = 0;
  for (int b = 0; b < 65536 && found < 2; ++b) {
    unsigned c = hist[b];
    while (found < 2 && (unsigned long long)ranks[found] < cum + c) {
      selhi[found] = (unsigned)b;
      selrem[found] = (unsigned)((unsigned long long)ranks[found] - cum);
      ++found;
    }
    cum += c;
  }
}
__global__ void hist_lo_kernel(const float* vals, const unsigned* selhi,
                               unsigned* hist2, int E) {
  int e = blockIdx.x * blockDim.x + threadIdx.x;
  if (e >= E) return;
  unsigned key = f2ord(vals[e]);
  unsigned hi = key >> 16, lo = key & 0xFFFFu;
  if (hi == selhi[0]) atomAddU(&hist2[lo], 1u);
  if (hi == selhi[1]) atomAddU(&hist2[65536u + lo], 1u);
}
__global__ void pick_lo_kernel(const unsigned* hist2, const unsigned* selhi,
                               const unsigned* selrem, float frac, float* thr) {
  float v[2];
  for (int j = 0; j < 2; ++j) {
    unsigned long long cum = 0;
    unsigned key = (selhi[j] << 16);
    for (int b = 0; b < 65536; ++b) {
      unsigned c = hist2[(size_t)j * 65536 + b];
      if ((unsigned long long)selrem[j] < cum + c) { key |= (unsigned)b; break; }
      cum += c;
    }
    v[j] = ord2f(key);
  }
  thr[0] = v[0] + frac * (v[1] - v[0]);
}

// ---------- SpMM scatter: fout += w[e] * u[col[e]] into row[e] ----------
__global__ void scatter_kernel(const float* __restrict__ u, const float* __restrict__ w,
                               const int* __restrict__ row, const int* __restrict__ col,
                               float* __restrict__ fout, int E) {
  int e = blockIdx.x * blockDim.x + threadIdx.x;
  if (e >= E) return;
  float we = w[e];
  if (we == 0.0f) return;
  const float4* uv = (const float4*)(u + (size_t)col[e] * 64);
  float* op = fout + (size_t)row[e] * 64;
  #pragma unroll
  for (int i = 0; i < 16; ++i) {
    float4 v = uv[i];
    atomAddF(op + i * 4 + 0, we * v.x);
    atomAddF(op + i * 4 + 1, we * v.y);
    atomAddF(op + i * 4 + 2, we * v.z);
    atomAddF(op + i * 4 + 3, we * v.w);
  }
}

// ---------- RK4 stage combine: k = fout - u; acc update; yt = y + c*k (or final) ----------
// No __restrict__: u==yt / y==yout alias by design (pure elementwise, same index).
__global__ void combine_kernel(const float* fout, const float* u, const float* y,
                               float* acc, float* yt, float* yout,
                               float accw, float ytc, float finw, int mode, size_t n) {
  size_t i = (size_t)blockIdx.x * blockDim.x + threadIdx.x;
  if (i >= n) return;
  float kv = fout[i] - u[i];
  if (mode == 0)      { acc[i] = kv;            yt[i] = y[i] + ytc * kv; }
  else if (mode == 1) { acc[i] += accw * kv;    yt[i] = y[i] + ytc * kv; }
  else                { yout[i] = y[i] + finw * (acc[i] + kv); }
}

// ---------- host driver ----------
extern "C" void kernel_launch(void* const* d_in, const int* in_sizes, int n_in,
                              void* d_out, int out_size, void* d_ws, size_t ws_size,
                              hipStream_t stream) {
  (void)n_in; (void)out_size; (void)ws_size;
  const float* x  = (const float*)d_in[0];
  const int*   ei = (const int*)d_in[1];      // jax default x64-off: int32
  const float* Wq = (const float*)d_in[2];
  const float* Wk = (const float*)d_in[3];

  const int N = in_sizes[0] / 64;
  const int E = in_sizes[1] / 2;
  const int* row = ei;
  const int* col = ei + E;
  const size_t ND = (size_t)N * 64;
  const int NH = N * 4;

  char* base = (char*)d_ws;
  size_t off = 0;
  auto alloc = [&](size_t bytes) -> void* {
    void* p = base + off;
    off += (bytes + 255) & ~(size_t)255;
    return p;
  };
  float*    q       = (float*)alloc(ND * 4);
  float*    kf      = (float*)alloc(ND * 4);
  float*    scores  = (float*)alloc((size_t)E * 4 * 4);
  unsigned* smax    = (unsigned*)alloc((size_t)NH * 4);
  float*    ssum    = (float*)alloc((size_t)NH * 4);
  float*    meanatt = (float*)alloc((size_t)E * 4);
  float*    w       = (float*)alloc((size_t)E * 4);
  unsigned* hist1   = (unsigned*)alloc(65536u * 4);
  unsigned* hist2   = (unsigned*)alloc(2u * 65536u * 4);
  unsigned* selhi   = (unsigned*)alloc(2 * 4);
  unsigned* selrem  = (unsigned*)alloc(2 * 4);
  float*    thr     = (float*)alloc(4);
  float*    yt      = (float*)alloc(ND * 4);
  float*    fout    = (float*)alloc(ND * 4);
  float*    accb    = (float*)alloc(ND * 4);

  float* y = (float*)d_out;

  const int gE  = (E + THREADS - 1) / THREADS;
  const int gND = (int)((ND + THREADS - 1) / THREADS);
  const int gNH = (NH + THREADS - 1) / THREADS;

  // 1) projections via WMMA f32
  int ntiles = N / 16;
  int gP = (ntiles * 32 + THREADS - 1) / THREADS;
  proj_wmma_kernel<<<gP, THREADS, 0, stream>>>(x, Wq, Wk, q, kf, ntiles);

  // 2) raw scores
  score_kernel<<<gE, THREADS, 0, stream>>>(q, kf, row, col, scores, E);

  // 3) pass-1 segment softmax -> mean_att0
  fill_u32_kernel<<<gNH, THREADS, 0, stream>>>(smax, 0u, (size_t)NH);
  fill_u32_kernel<<<gNH, THREADS, 0, stream>>>((unsigned*)ssum, 0u, (size_t)NH);
  segmax_kernel<<<gE, THREADS, 0, stream>>>(scores, row, smax, nullptr, nullptr, 0, E);
  expsum_kernel<<<gE, THREADS, 0, stream>>>(scores, smax, row, ssum, nullptr, nullptr, 0, E);
  attw_kernel<<<gE, THREADS, 0, stream>>>(scores, smax, ssum, row, nullptr, nullptr, 0, meanatt, E);

  // 4) device-side 5% quantile (linear interpolation between ranks)
  double pos = 0.05 * (double)(E - 1);
  unsigned r0 = (unsigned)pos;
  unsigned r1 = (r0 + 1 < (unsigned)E) ? r0 + 1 : r0;
  float frac = (float)(pos - (double)r0);
  fill_u32_kernel<<<(65536 + THREADS - 1) / THREADS, THREADS, 0, stream>>>(hist1, 0u, (size_t)65536);
  hist_hi_kernel<<<gE, THREADS, 0, stream>>>(meanatt, hist1, E);
  pick_hi_kernel<<<1, 1, 0, stream>>>(hist1, r0, r1, selhi, selrem);
  fill_u32_kernel<<<(2 * 65536 + THREADS - 1) / THREADS, THREADS, 0, stream>>>(hist2, 0u, (size_t)(2 * 65536));
  hist_lo_kernel<<<gE, THREADS, 0, stream>>>(meanatt, selhi, hist2, E);
  pick_lo_kernel<<<1, 1, 0, stream>>>(hist2, selhi, selrem, frac, thr);

  // 5) pass-2 masked segment softmax -> edge weights w
  fill_u32_kernel<<<gNH, THREADS, 0, stream>>>(smax, 0u, (size_t)NH);
  fill_u32_kernel<<<gNH, THREADS, 0, stream>>>((unsigned*)ssum, 0u, (size_t)NH);
  segmax_kernel<<<gE, THREADS, 0, stream>>>(scores, row, smax, meanatt, thr, 1, E);
  expsum_kernel<<<gE, THREADS, 0, stream>>>(scores, smax, row, ssum, meanatt, thr, 1, E);
  attw_kernel<<<gE, THREADS, 0, stream>>>(scores, smax, ssum, row, meanatt, thr, 1, w, E);

  // 6) RK4 on dx/dt = A x - x, 4 steps, dt = 0.25; y lives in d_out
  copy_f32_kernel<<<gND, THREADS, 0, stream>>>(y, x, ND);
  const float dt = 0.25f;
  for (int s = 0; s < 4; ++s) {
    // stage 1: u = y
    fill_u32_kernel<<<gND, THREADS, 0, stream>>>((unsigned*)fout, 0u, ND);
    scatter_kernel<<<gE, THREADS, 0, stream>>>(y, w, row, col, fout, E);
    combine_kernel<<<gND, THREADS, 0, stream>>>(fout, y, y, accb, yt, y, 1.0f, 0.5f * dt, 0.0f, 0, ND);
    // stage 2: u = yt
    fill_u32_kernel<<<gND, THREADS, 0, stream>>>((unsigned*)fout, 0u, ND);
    scatter_kernel<<<gE, THREADS, 0, stream>>>(yt, w, row, col, fout, E);
    combine_kernel<<<gND, THREADS, 0, stream>>>(fout, yt, y, accb, yt, y, 2.0f, 0.5f * dt, 0.0f, 1, ND);
    // stage 3: u = yt
    fill_u32_kernel<<<gND, THREADS, 0, stream>>>((unsigned*)fout, 0u, ND);
    scatter_kernel<<<gE, THREADS, 0, stream>>>(yt, w, row, col, fout, E);
    combine_kernel<<<gND, THREADS, 0, stream>>>(fout, yt, y, accb, yt, y, 2.0f, dt, 0.0f, 1, ND);
    // stage 4: u = yt, final: y += dt/6 * (acc + k4)
    fill_u32_kernel<<<gND, THREADS, 0, stream>>>((unsigned*)fout, 0u, ND);
    scatter_kernel<<<gE, THREADS, 0, stream>>>(yt, w, row, col, fout, E);
    combine_kernel<<<gND, THREADS, 0, stream>>>(fout, yt, y, accb, yt, y, 0.0f, 0.0f, dt / 6.0f, 2, ND);
  }
}